// Voxelizer_35338990912022
// MI455X (gfx1250) — compile-verified
//
#include <hip/hip_runtime.h>

// Problem constants (match reference: RES=256, B=1, C=8)
#define RES        256
#define CH         8
#define HBITS      18
#define HSIZE      (1u << HBITS)          // 262144 slots, ~38% load for 100k pts
#define HMASK      (HSIZE - 1u)
#define IMG_ELEMS  (3 * CH * RES * RES)   // 1,572,864 output image elements
#define MAP_NEG_INF 0x007FFFFFu           // monotonic mapping of -inf

// Monotonic float -> u32 mapping: preserves total order under unsigned compare.
__device__ __forceinline__ unsigned mapf(float f) {
    int s = __float_as_int(f);
    return (unsigned)(s ^ ((s >> 31) | 0x80000000));
}
__device__ __forceinline__ float unmapf(unsigned u) {
    int bits = (u & 0x80000000u) ? (int)(u ^ 0x80000000u) : (int)~u;
    return __int_as_float(bits);
}

// gfx1250 async copy: global (SADDR base + 32-bit VGPR offset) -> LDS (byte
// address in VGPR). Tracked by ASYNCcnt; per-wave wait, no barrier needed when
// each lane consumes only its own 16B.
__device__ __forceinline__ void async_g2l_b128(unsigned lds_byte_addr,
                                               unsigned gbyte_off,
                                               const void* gbase) {
    asm volatile("global_load_async_to_lds_b128 %0, %1, %2"
                 :
                 : "v"(lds_byte_addr), "v"(gbyte_off), "s"(gbase)
                 : "memory");
}
__device__ __forceinline__ void wait_asynccnt0() {
    asm volatile("s_wait_asynccnt 0x0" ::: "memory");
}

// ---------------------------------------------------------------------------
// 1) Re-init sparse workspace every call (harness poisons ws). b128 stores.
//    Launched with HSIZE*CH/4 threads (largest region / 4).
// ---------------------------------------------------------------------------
__global__ void vox_init(int4* __restrict__ keys, int4* __restrict__ cnts,
                         float4* __restrict__ sums, uint4* __restrict__ mmax,
                         int4* __restrict__ ccnt) {
    unsigned i = blockIdx.x * blockDim.x + threadIdx.x;
    if (i < HSIZE / 4) {
        keys[i] = make_int4(-1, -1, -1, -1);
        cnts[i] = make_int4(0, 0, 0, 0);
    }
    if (i < HSIZE * CH / 4)
        sums[i] = make_float4(0.f, 0.f, 0.f, 0.f);
    if (i < IMG_ELEMS / 4)
        mmax[i] = make_uint4(MAP_NEG_INF, MAP_NEG_INF, MAP_NEG_INF, MAP_NEG_INF);
    if (i < 3 * RES * RES / 4)
        ccnt[i] = make_int4(0, 0, 0, 0);
}

// ---------------------------------------------------------------------------
// 2) Scatter points into an open-addressing hash table keyed by voxel id.
//    global_atomic_cmpswap for slot claim, global_atomic_add_f32 for sums.
// ---------------------------------------------------------------------------
__global__ void vox_scatter(const float4* __restrict__ coords,
                            const float4* __restrict__ feats,
                            int* __restrict__ keys, int* __restrict__ cnts,
                            float* __restrict__ sums, int n) {
    int i = blockIdx.x * blockDim.x + threadIdx.x;
    if (i >= n) return;

    __builtin_prefetch(&feats[2 * i], 0, 0);   // global_prefetch_b8 on feature row

    float4 co = coords[i];                     // global_load_b128: x,y,z,batch
    int x = min(max((int)floorf(co.x * (float)RES), 0), RES - 1);
    int y = min(max((int)floorf(co.y * (float)RES), 0), RES - 1);
    int z = min(max((int)floorf(co.z * (float)RES), 0), RES - 1);
    int b = (int)co.w;                         // B==1 -> 0
    int lin = ((b * RES + x) * RES + y) * RES + z;  // < 2^24, fits int

    // Integer finalizer hash -> slot, linear probing.
    unsigned h = (unsigned)lin;
    h ^= h >> 16; h *= 0x45d9f3bu; h ^= h >> 16;
    h &= HMASK;
    for (;;) {
        int prev = atomicCAS(&keys[h], -1, lin);
        if (prev == -1 || prev == lin) break;
        h = (h + 1) & HMASK;
    }

    atomicAdd(&cnts[h], 1);
    float4 f0 = feats[2 * i];                  // global_load_b128
    float4 f1 = feats[2 * i + 1];
    float* s = sums + (size_t)h * CH;
    atomicAdd(s + 0, f0.x); atomicAdd(s + 1, f0.y);
    atomicAdd(s + 2, f0.z); atomicAdd(s + 3, f0.w);
    atomicAdd(s + 4, f1.x); atomicAdd(s + 5, f1.y);
    atomicAdd(s + 6, f1.z); atomicAdd(s + 7, f1.w);
}

// ---------------------------------------------------------------------------
// 3) For each occupied voxel: mean = sum/cnt, then atomicMax (on monotonic
//    u32 mapping) into the three projection images; count column occupancy.
//    Image layout == output layout: idx = ((view*CH + c) << 16) | (p << 8) | q
//      view 0 (max over H=x): pixel (p=y, q=z)
//      view 1 (max over W=y): pixel (p=x, q=z)
//      view 2 (max over Z=z): pixel (p=x, q=y)
// ---------------------------------------------------------------------------
__global__ void vox_project(const int* __restrict__ keys,
                            const int* __restrict__ cnts,
                            const float* __restrict__ sums,
                            unsigned* __restrict__ mmax,
                            int* __restrict__ ccnt) {
    unsigned i = blockIdx.x * blockDim.x + threadIdx.x;
    if (i >= HSIZE) return;
    int k = keys[i];
    if (k < 0) return;

    float inv = 1.0f / (float)cnts[i];
    int z = k & 255, y = (k >> 8) & 255, x = (k >> 16) & 255;
    int pH = (y << 8) | z;
    int pW = (x << 8) | z;
    int pZ = (x << 8) | y;

    const float* s = sums + (size_t)i * CH;
#pragma unroll
    for (int c = 0; c < CH; ++c) {
        unsigned u = mapf(s[c] * inv);
        atomicMax(&mmax[(unsigned)(((0 * CH + c) << 16) | pH)], u);
        atomicMax(&mmax[(unsigned)(((1 * CH + c) << 16) | pW)], u);
        atomicMax(&mmax[(unsigned)(((2 * CH + c) << 16) | pZ)], u);
    }
    atomicAdd(&ccnt[0 * RES * RES + pH], 1);
    atomicAdd(&ccnt[1 * RES * RES + pW], 1);
    atomicAdd(&ccnt[2 * RES * RES + pZ], 1);
}

// ---------------------------------------------------------------------------
// 4) Tiled streaming finalize using gfx1250 async global->LDS copies.
//    Each block handles 1024 contiguous outputs (256 threads x 4). A tile
//    never crosses a (view,channel) plane (65536 % 1024 == 0), so its ccnt
//    region is contiguous. Each lane async-loads its own 16B of mmax and
//    16B of ccnt into LDS, waits on its wave's ASYNCcnt, reads back via
//    ds_load_b128, then writes a float4.
// ---------------------------------------------------------------------------
__global__ void vox_finalize(const unsigned* __restrict__ mmax,
                             const int* __restrict__ ccnt,
                             const int* __restrict__ box_class,
                             float* __restrict__ out, int out_size) {
    __shared__ unsigned lds_buf[2048];         // [0..1023] mmax, [1024..2047] ccnt

    const int i0 = blockIdx.x * 1024 + threadIdx.x * 4;   // first of 4 outputs
    const int v  = i0 >> 19;                   // view id (plane-constant per tile)
    const int pq = i0 & 0xFFFF;                // pixel (p<<8)|q of first element

    const unsigned lds_base = (unsigned)(uintptr_t)lds_buf;  // LDS byte offset
    const unsigned mo = lds_base + threadIdx.x * 16u;
    const unsigned co = lds_base + 4096u + threadIdx.x * 16u;

    async_g2l_b128(mo, (unsigned)i0 * 4u, mmax);
    async_g2l_b128(co, (unsigned)(v * RES * RES + pq) * 4u, ccnt);
    wait_asynccnt0();

    const uint4 mm = *(const uint4*)&lds_buf[threadIdx.x * 4];        // ds_load_b128
    const uint4 cc = *(const uint4*)&lds_buf[1024 + threadIdx.x * 4]; // ds_load_b128

    float4 r;
    {
        float a = unmapf(mm.x); if ((int)cc.x < RES) a = fmaxf(a, 0.0f); r.x = a;
        float b = unmapf(mm.y); if ((int)cc.y < RES) b = fmaxf(b, 0.0f); r.y = b;
        float c = unmapf(mm.z); if ((int)cc.z < RES) c = fmaxf(c, 0.0f); r.z = c;
        float d = unmapf(mm.w); if ((int)cc.w < RES) d = fmaxf(d, 0.0f); r.w = d;
    }
    *(float4*)(out + i0) = r;                  // global_store_b128

    if (blockIdx.x == 0 && threadIdx.x < 3 && (IMG_ELEMS + (int)threadIdx.x) < out_size) {
        out[IMG_ELEMS + threadIdx.x] = (float)box_class[0];  // img_class repeated 3x
    }
}

// ---------------------------------------------------------------------------
extern "C" void kernel_launch(void* const* d_in, const int* in_sizes, int n_in,
                              void* d_out, int out_size, void* d_ws, size_t ws_size,
                              hipStream_t stream) {
    const float4* coords    = (const float4*)d_in[0];   // (N,4) f32
    const float4* feats     = (const float4*)d_in[1];   // (N,8) f32
    const int*    box_class = (const int*)d_in[2];      // (B,)  i32
    const int n = in_sizes[0] / 4;

    // Workspace layout (~17.6 MB total; all offsets 256B-aligned):
    char* ws = (char*)d_ws;
    size_t off = 0;
    int*      keys = (int*)(ws + off);      off += (size_t)HSIZE * sizeof(int);
    int*      cnts = (int*)(ws + off);      off += (size_t)HSIZE * sizeof(int);
    float*    sums = (float*)(ws + off);    off += (size_t)HSIZE * CH * sizeof(float);
    unsigned* mmax = (unsigned*)(ws + off); off += (size_t)IMG_ELEMS * sizeof(unsigned);
    int*      ccnt = (int*)(ws + off);      off += (size_t)(3 * RES * RES) * sizeof(int);
    (void)ws_size; (void)n_in; (void)off;

    const dim3 blk(256);
    vox_init    <<<(HSIZE * CH / 4 + 255) / 256, blk, 0, stream>>>(
        (int4*)keys, (int4*)cnts, (float4*)sums, (uint4*)mmax, (int4*)ccnt);
    vox_scatter <<<(n + 255) / 256, blk, 0, stream>>>(coords, feats, keys, cnts, sums, n);
    vox_project <<<HSIZE / 256,     blk, 0, stream>>>(keys, cnts, sums, mmax, ccnt);
    vox_finalize<<<IMG_ELEMS / 1024, blk, 0, stream>>>(mmax, ccnt, box_class,
                                                       (float*)d_out, out_size);
}